// DeepGate3_62070867362293
// MI455X (gfx1250) — compile-verified
//
#include <hip/hip_runtime.h>
#include <hip/hip_bf16.h>

typedef __attribute__((ext_vector_type(2))) float v2f;
typedef __attribute__((ext_vector_type(8))) float v8f;

#define DIM    128
#define DFF    512
#define TTW    64
#define PMAX   8
#define NTOK   9
#define TILE_M 16
#define MB     2                       // M sub-tiles per block
#define HOPS_PER_BLOCK (MB * TILE_M)   // 32 hops / block

// LDS row strides in floats, padded so WMMA A-operand reads (bank =
// (row*stride + k) % 64) hit 16 distinct banks: stride % 64 == 4.
#define AS 132   // pooled A tile  [32][128] -> 32*132*4 = 16896 B
#define HS 516   // hidden H tile  [32][512] -> 32*516*4 = 66048 B
// Ptile: 2 K-halves x 32 rows x 64 -> 16384 B.  Total LDS ~= 99 KB.

__global__ __launch_bounds__(256, 2)
void deepgate3_fused(const float* __restrict__ hf,
                     const float* __restrict__ w_q,
                     const float* __restrict__ W1,
                     const float* __restrict__ b1,
                     const float* __restrict__ gamma,
                     const float* __restrict__ beta,
                     const float* __restrict__ W2,
                     const float* __restrict__ b2,
                     const int*   __restrict__ hop_pi,
                     const int*   __restrict__ hop_pi_stats,
                     const int*   __restrict__ hop_po,
                     float*       __restrict__ out,
                     int n_hops)
{
    __shared__ float Atile[HOPS_PER_BLOCK * AS];
    __shared__ float Htile[HOPS_PER_BLOCK * HS];
    __shared__ float Ptile[2 * HOPS_PER_BLOCK * TTW];

    const int tid   = threadIdx.x;
    const int lane  = tid & 31;
    const int wave  = tid >> 5;
    const int khalf = lane >> 4;   // K half-select per WMMA f32 16x16x4 layout
    const int ml    = lane & 15;   // M (A) / N (B,C) lane index

    const int hop_base = blockIdx.x * HOPS_PER_BLOCK;

    // ---------------- Phase A: masked attention pooling (4 hops / wave) ----
    const int d0 = lane * 4;
    const float4 wq4 = *(const float4*)(w_q + d0);

    for (int hh = 0; hh < HOPS_PER_BLOCK / 8; ++hh) {
        const int hl = wave * (HOPS_PER_BLOCK / 8) + hh;    // local hop 0..31
        int h = hop_base + hl;
        if (h >= n_hops) h = n_hops - 1;                    // clamp (no divergence)

        float4 tok[NTOK];
        float  sc[NTOK];
        bool   vld[NTOK];
        for (int p = 0; p < PMAX; ++p) {
            const int idx = hop_pi[h * PMAX + p];
            const int st  = hop_pi_stats[h * PMAX + p];
            const float4 t = *(const float4*)(hf + (size_t)idx * DIM + d0);
            tok[p] = t;
            sc[p]  = t.x * wq4.x + t.y * wq4.y + t.z * wq4.z + t.w * wq4.w;
            vld[p] = (st != -1);
        }
        {
            const int idx = hop_po[h];
            const float4 t = *(const float4*)(hf + (size_t)idx * DIM + d0);
            tok[PMAX] = t;
            sc[PMAX]  = t.x * wq4.x + t.y * wq4.y + t.z * wq4.z + t.w * wq4.w;
            vld[PMAX] = true;
        }
        // butterfly reduce the 9 per-lane dot partials across wave32
        for (int off = 16; off > 0; off >>= 1)
            for (int p = 0; p < NTOK; ++p)
                sc[p] += __shfl_xor(sc[p], off, 32);

        const float inv_sqrt_d = 0.088388347648318447f;     // 1/sqrt(128)
        float mx = -1e30f;
        for (int p = 0; p < NTOK; ++p) {
            sc[p] = vld[p] ? sc[p] * inv_sqrt_d : -1e30f;
            mx = fmaxf(mx, sc[p]);
        }
        float se = 0.f;
        for (int p = 0; p < NTOK; ++p) { sc[p] = expf(sc[p] - mx); se += sc[p]; }
        const float inv = 1.f / se;

        float4 pooled = make_float4(0.f, 0.f, 0.f, 0.f);
        for (int p = 0; p < NTOK; ++p) {
            const float a = sc[p] * inv;
            pooled.x += a * tok[p].x;  pooled.y += a * tok[p].y;
            pooled.z += a * tok[p].z;  pooled.w += a * tok[p].w;
        }
        *(float4*)(&Atile[hl * AS + d0]) = pooled;          // 16B aligned (AS=132)
    }
    __syncthreads();

    // ---------------- Phase B: GEMM1 32x128 @ 128x512 (+b1, ReLU) ----------
    // fp32 WMMA layouts: A 16x4 (lanes0-15: K=k,k+1; lanes16-31: K=k+2,k+3),
    // B 4x16 (VGPR j: K=k+2*khalf+j, N=ml), C 16x16 (VGPR i: M=i+8*khalf, N=ml).
    // One B fragment feeds MB WMMAs (register reuse across M sub-tiles).
    for (int t = 0; t < 4; ++t) {
        const int n0 = (wave * 4 + t) * 16;
        const int n  = n0 + ml;
        v8f acc[MB];
        for (int mt = 0; mt < MB; ++mt)
            acc[mt] = (v8f){0.f, 0.f, 0.f, 0.f, 0.f, 0.f, 0.f, 0.f};
        for (int k = 0; k < DIM; k += 4) {
            const int kk = k + khalf * 2;
            v2f b;
            b.x = W1[(size_t)kk * DFF + n];
            b.y = W1[(size_t)(kk + 1) * DFF + n];
            for (int mt = 0; mt < MB; ++mt) {
                v2f a;
                a.x = Atile[(mt * TILE_M + ml) * AS + kk];
                a.y = Atile[(mt * TILE_M + ml) * AS + kk + 1];
                acc[mt] = __builtin_amdgcn_wmma_f32_16x16x4_f32(
                              false, a, false, b, (short)0, acc[mt], false, false);
            }
        }
        const float bias = b1[n];
        for (int mt = 0; mt < MB; ++mt)
            for (int i = 0; i < 8; ++i) {
                const int m = mt * TILE_M + i + khalf * 8;
                float v = acc[mt][i] + bias;
                Htile[m * HS + n] = v > 0.f ? v : 0.f;
            }
    }
    __syncthreads();

    // ---------------- Phase C: LayerNorm(512) per hop row (4 rows / wave) --
    for (int hh = 0; hh < HOPS_PER_BLOCK / 8; ++hh) {
        const int r = wave * (HOPS_PER_BLOCK / 8) + hh;
        float s = 0.f, s2 = 0.f;
        for (int c = lane; c < DFF; c += 32) {
            const float x = Htile[r * HS + c];
            s += x; s2 += x * x;
        }
        for (int off = 16; off > 0; off >>= 1) {
            s  += __shfl_xor(s,  off, 32);
            s2 += __shfl_xor(s2, off, 32);
        }
        const float mu   = s * (1.f / DFF);
        const float var  = s2 * (1.f / DFF) - mu * mu;      // population var (jnp.var)
        const float rstd = rsqrtf(var + 1e-5f);
        for (int c = lane; c < DFF; c += 32) {
            const float x = Htile[r * HS + c];
            Htile[r * HS + c] = (x - mu) * rstd * gamma[c] + beta[c];
        }
    }
    __syncthreads();

    // ---------------- Phase D: GEMM2 32x512 @ 512x64, K split over waves ---
    {
        const int nt = wave & 3;          // N tile 0..3
        const int kh = wave >> 2;         // K half 0..1
        const int n  = nt * 16 + ml;
        const int kbeg = kh * 256;
        v8f acc[MB];
        for (int mt = 0; mt < MB; ++mt)
            acc[mt] = (v8f){0.f, 0.f, 0.f, 0.f, 0.f, 0.f, 0.f, 0.f};
        for (int k = kbeg; k < kbeg + 256; k += 4) {
            const int kk = k + khalf * 2;
            v2f b;
            b.x = W2[(size_t)kk * TTW + n];
            b.y = W2[(size_t)(kk + 1) * TTW + n];
            for (int mt = 0; mt < MB; ++mt) {
                v2f a;
                a.x = Htile[(mt * TILE_M + ml) * HS + kk];
                a.y = Htile[(mt * TILE_M + ml) * HS + kk + 1];
                acc[mt] = __builtin_amdgcn_wmma_f32_16x16x4_f32(
                              false, a, false, b, (short)0, acc[mt], false, false);
            }
        }
        for (int mt = 0; mt < MB; ++mt)
            for (int i = 0; i < 8; ++i) {
                const int m = mt * TILE_M + i + khalf * 8;
                Ptile[(kh * HOPS_PER_BLOCK + m) * TTW + n] = acc[mt][i];
            }
    }
    __syncthreads();

    // ---------------- Phase E: combine K halves, +b2, store ----------------
    for (int e = tid; e < HOPS_PER_BLOCK * TTW; e += 256) {
        const int r = e >> 6;
        const int c = e & 63;
        const int h = hop_base + r;
        if (h < n_hops) {
            out[(size_t)h * TTW + c] =
                Ptile[r * TTW + c] + Ptile[(HOPS_PER_BLOCK + r) * TTW + c] + b2[c];
        }
    }
}

extern "C" void kernel_launch(void* const* d_in, const int* in_sizes, int n_in,
                              void* d_out, int out_size, void* d_ws, size_t ws_size,
                              hipStream_t stream) {
    (void)n_in; (void)out_size; (void)d_ws; (void)ws_size;
    const float* hf    = (const float*)d_in[0];
    const float* w_q   = (const float*)d_in[1];
    const float* W1    = (const float*)d_in[2];
    const float* b1    = (const float*)d_in[3];
    const float* gamma = (const float*)d_in[4];
    const float* beta  = (const float*)d_in[5];
    const float* W2    = (const float*)d_in[6];
    const float* b2    = (const float*)d_in[7];
    const int* hop_pi       = (const int*)d_in[8];
    const int* hop_pi_stats = (const int*)d_in[9];
    const int* hop_po       = (const int*)d_in[10];
    const int n_hops = in_sizes[10];

    const int n_blocks = (n_hops + HOPS_PER_BLOCK - 1) / HOPS_PER_BLOCK;
    deepgate3_fused<<<n_blocks, 256, 0, stream>>>(
        hf, w_q, W1, b1, gamma, beta, W2, b2,
        hop_pi, hop_pi_stats, hop_po, (float*)d_out, n_hops);
}